// KmeansEuclideanObjectTokens_1013612282358
// MI455X (gfx1250) — compile-verified
//
#include <hip/hip_runtime.h>

// MI455X / gfx1250, wave32. Batched k-means with V_WMMA_F32_16X16X4_F32 for the
// distance GEMM. One workgroup per batch; x (128MB) is L2-resident across iters.

typedef __attribute__((ext_vector_type(2))) float v2f;
typedef __attribute__((ext_vector_type(8))) float v8f;

#define BATCHES   32
#define NPTS      4096
#define CH        256
#define KCLUST    16
#define MAXITERS  20
#define TOLERANCE 1e-6f

__global__ __launch_bounds__(256)
void kmeans_wmma_kernel(const float* __restrict__ x,
                        const int* __restrict__ init_idx,
                        float* __restrict__ out) {
  __shared__ float c_lds[KCLUST][CH];     // current centroids, 16KB
  __shared__ float sum_lds[KCLUST][CH];   // per-cluster accumulators, 16KB
  __shared__ float cnorm[KCLUST];
  __shared__ int   counts[KCLUST];
  __shared__ int   err_bits;

  const int b    = blockIdx.x;
  const int tid  = threadIdx.x;
  const int wave = tid >> 5;
  const int lane = tid & 31;
  const int col  = lane & 15;   // cluster column held by this lane in B/C/D frags
  const int half = lane >> 4;   // 0: K=0,1 / rows 0-7 ; 1: K=2,3 / rows 8-15

  const float* xb = x + (size_t)b * NPTS * CH;

  // --- init centroids: c0 = x[b, init_idx, :] ---
  for (int e = tid; e < KCLUST * CH; e += 256) {
    int k = e >> 8, ch = e & 255;
    c_lds[k][ch] = xb[(size_t)init_idx[k] * CH + ch];
  }
  __syncthreads();

  for (int iter = 0; iter < MAXITERS; ++iter) {
    // --- zero accumulators ---
    for (int e = tid; e < KCLUST * CH; e += 256) sum_lds[e >> 8][e & 255] = 0.0f;
    if (tid < KCLUST) { counts[tid] = 0; cnorm[tid] = 0.0f; }
    if (tid == 0) err_bits = 0;
    __syncthreads();

    // --- cnorm[k] = sum_ch c[k][ch]^2 (16 threads per cluster) ---
    {
      int k = tid >> 4, p = tid & 15;
      float s = 0.0f;
      for (int ch = p; ch < CH; ch += 16) { float v = c_lds[k][ch]; s += v * v; }
      atomicAdd(&cnorm[k], s);
    }
    __syncthreads();

    // --- assignment (WMMA distance tiles) + scatter accumulation ---
    for (int tile = wave; tile < NPTS / 16; tile += 8) {
      const float* xt = xb + (size_t)tile * 16 * CH;

      // dot tile: A = x rows (16x4 f32 frag), B = c^T (4x16 frag from LDS)
      v8f acc = {};
      const float* arow = xt + (size_t)col * CH;
      for (int kk = 0; kk < CH; kk += 4) {
        int off = kk + half * 2;
        v2f a;  a.x  = arow[off];          a.y  = arow[off + 1];
        v2f bb; bb.x = c_lds[col][off];    bb.y = c_lds[col][off + 1];
        acc = __builtin_amdgcn_wmma_f32_16x16x4_f32(
            /*neg_a=*/false, a, /*neg_b=*/false, bb,
            /*c_mod=*/(short)0, acc, /*reuse_a=*/false, /*reuse_b=*/false);
      }

      // distances d = ||c||^2 - 2*dot ; argmin over 16 cluster columns, which
      // live across the 16 lanes of each wave32 half. Lowest-index tie-break.
      int best[8];
      float cn = cnorm[col];
      #pragma unroll
      for (int r = 0; r < 8; ++r) {
        float d = cn - 2.0f * acc[r];
        int idx = col;
        #pragma unroll
        for (int m = 8; m >= 1; m >>= 1) {
          float od = __shfl_xor(d, m, 32);
          int   oi = __shfl_xor(idx, m, 32);
          if (od < d || (od == d && oi < idx)) { d = od; idx = oi; }
        }
        best[r] = idx;   // row (r + 8*half) of this tile -> cluster idx
      }

      // cooperative scatter-add of each of the 16 rows into its cluster sum
      #pragma unroll 1
      for (int rg = 0; rg < 16; ++rg) {
        int src = (rg < 8) ? 0 : 16;                 // half holding this row
        int k = __shfl(best[rg & 7], src, 32);       // broadcast to whole wave
        const float* xr = xt + (size_t)rg * CH;
        #pragma unroll
        for (int ch = lane; ch < CH; ch += 32)
          atomicAdd(&sum_lds[k][ch], xr[ch]);        // ds_add_f32
        if (lane == 0) atomicAdd(&counts[k], 1);
      }
    }
    __syncthreads();

    // --- centroid update + convergence (max |c_new - c_old|) ---
    for (int e = tid; e < KCLUST * CH; e += 256) {
      int k = e >> 8, ch = e & 255;
      int cnt = counts[k];
      float oldv = c_lds[k][ch];
      float newv = (cnt > 0) ? (sum_lds[k][ch] / (float)cnt) : oldv;
      float diff = fabsf(newv - oldv);
      atomicMax(&err_bits, __float_as_int(diff));    // valid: diff >= 0
      c_lds[k][ch] = newv;                           // each entry owned by one thread
    }
    __syncthreads();

    float err = __int_as_float(err_bits);            // uniform across block
    if (err < TOLERANCE) break;                      // uniform branch
    __syncthreads();                                 // protect err_bits before re-zero
  }

  // --- emit centroids: out[b, k, ch] ---
  float* ob = out + (size_t)b * KCLUST * CH;
  for (int e = tid; e < KCLUST * CH; e += 256)
    ob[e] = c_lds[e >> 8][e & 255];
}

extern "C" void kernel_launch(void* const* d_in, const int* in_sizes, int n_in,
                              void* d_out, int out_size, void* d_ws, size_t ws_size,
                              hipStream_t stream) {
  (void)in_sizes; (void)n_in; (void)out_size; (void)d_ws; (void)ws_size;
  const float* x        = (const float*)d_in[0];
  const int*   init_idx = (const int*)d_in[1];
  float*       out      = (float*)d_out;
  kmeans_wmma_kernel<<<BATCHES, 256, 0, stream>>>(x, init_idx, out);
}